// StockPredictionModel_39977555591206
// MI455X (gfx1250) — compile-verified
//
#include <hip/hip_runtime.h>
#include <cstdint>

// ---------------------------------------------------------------------------
// Model dims (fixed by the reference)
// ---------------------------------------------------------------------------
constexpr int N_ = 4096, T_ = 128, F_ = 64, D_ = 64, T2_ = 64;
constexpr int E_ = 131072, L_ = 32, NH_ = 4, HD_ = 16, KC_ = 5;

typedef __attribute__((ext_vector_type(2))) float v2f;
typedef __attribute__((ext_vector_type(4))) float f4;
typedef __attribute__((ext_vector_type(8))) float v8f;

// ---------------------------------------------------------------------------
// CDNA5 WMMA: D(16x16,f32) = A(16x4,f32) x B(4x16,f32) + C   (wave32)
// A frag: lane p=l&15,h=l>>4 -> a[v] = A[p][2h+v] (per 4-wide K chunk)
// B frag: b[v] = B[2h+v][p];  C/D: c[j] = C[j+8h][p]
// ---------------------------------------------------------------------------
__device__ inline v8f wmma4(v2f a, v2f b, v8f c) {
  return __builtin_amdgcn_wmma_f32_16x16x4_f32(
      /*neg_a=*/false, a, /*neg_b=*/false, b,
      /*c_mod=*/(short)0, c, /*reuse_a=*/false, /*reuse_b=*/false);
}
__device__ inline v8f splat8(float x) {
  v8f r;
#pragma unroll
  for (int j = 0; j < 8; ++j) r[j] = x;
  return r;
}
__device__ inline v2f ldfrag(const float* p) { v2f r; r[0] = p[0]; r[1] = p[1]; return r; }

// ---------------------------------------------------------------------------
// Async global->LDS copy (CDNA5 TDM-lite path, tracked by ASYNCcnt).
// VFLAT async encoding: VDST = LDS byte-offset VGPR, VADDR = 64-bit global
// address (GV mode, saddr=off). Low 32 bits of a generic LDS pointer are the
// LDS byte offset (shared aperture base has zero low bits).
// ---------------------------------------------------------------------------
__device__ inline void async_ld_b128(uint32_t lds_off, const float* gptr) {
  asm volatile("global_load_async_to_lds_b128 %0, %1, off"
               :
               : "v"(lds_off), "v"((uint64_t)(uintptr_t)gptr)
               : "memory");
}
__device__ inline void wait_async0() {
  asm volatile("s_wait_asynccnt 0x0" ::: "memory");
}

#define WPAD 66   // weight LDS row stride (bank-conflict-free b64 frag loads)
#define XPAD 68   // activation LDS row stride (16B-aligned float4 staging)

// ---------------------------------------------------------------------------
// Fused GRU layer: whole T-step scan in one kernel. 4 waves/block, 16 rows/wave.
// Dead code removed: size-1 softmax => r == Wv x + bv  (Wq/Wk never used).
//   z  = sigmoid(Wz x + Uz h + (bWz+bUz))
//   ht = tanh(Wh x + bWh + (Wv x + bWv) * (Uh h + bUh))
//   h  = (1-z) h + z ht
// x tiles are double-buffered in LDS and fetched one step ahead with async
// B128 copies, overlapping the next step's HBM latency with this step's WMMAs.
// ---------------------------------------------------------------------------
__global__ void __launch_bounds__(128, 1) gru_layer_kernel(
    const float* __restrict__ in, long rowStride, long tStride, int Tsteps,
    const float* __restrict__ Wz, const float* __restrict__ bWz,
    const float* __restrict__ Uz, const float* __restrict__ bUz,
    const float* __restrict__ Wv, const float* __restrict__ bWv,
    const float* __restrict__ Wh, const float* __restrict__ bWh,
    const float* __restrict__ Uh, const float* __restrict__ bUh,
    float* __restrict__ seq_out, float* __restrict__ final_out) {
  extern __shared__ float smem[];
  float* sWz = smem;
  float* sUz = sWz + 64 * WPAD;
  float* sWv = sUz + 64 * WPAD;
  float* sWh = sWv + 64 * WPAD;
  float* sUh = sWh + 64 * WPAD;
  float* sbz = sUh + 64 * WPAD;  // bWz + bUz combined
  float* sbv = sbz + 64;
  float* sbh = sbv + 64;
  float* sbu = sbh + 64;
  float* sx = sbu + 64;               // [2 buf][4 waves][16][XPAD]
  float* sh = sx + 2 * 4 * 16 * XPAD; // [2 parity][4 waves][16][XPAD]

  const int tid = threadIdx.x;
  for (int i = tid; i < 64 * 64; i += blockDim.x) {
    int o = i >> 6, k = i & 63;
    sWz[o * WPAD + k] = Wz[i];
    sUz[o * WPAD + k] = Uz[i];
    sWv[o * WPAD + k] = Wv[i];
    sWh[o * WPAD + k] = Wh[i];
    sUh[o * WPAD + k] = Uh[i];
  }
  for (int i = tid; i < 64; i += blockDim.x) {
    sbz[i] = bWz[i] + bUz[i];
    sbv[i] = bWv[i];
    sbh[i] = bWh[i];
    sbu[i] = bUh[i];
  }
  const int wv = tid >> 5, lane = tid & 31;
  float* wx0 = sx + wv * 16 * XPAD;
  float* wx1 = sx + (4 + wv) * 16 * XPAD;
  float* wh0 = sh + wv * 16 * XPAD;
  float* wh1 = sh + (4 + wv) * 16 * XPAD;
  for (int i = lane; i < 16 * XPAD; i += 32) wh0[i] = 0.f;
  __syncthreads();

  const int p = lane & 15, half = lane >> 4;
  const long baseRow = (long)blockIdx.x * 64 + wv * 16;

  // per-lane staging pattern: idx = lane + 32*i -> (row, 16B column chunk)
  auto issue_x = [&](int step, float* buf) {
    const float* xin = in + baseRow * rowStride + (long)step * tStride;
    const uint32_t lbase = (uint32_t)(uintptr_t)buf;
#pragma unroll
    for (int i = 0; i < 8; ++i) {
      int idx = lane + 32 * i;
      int r = idx >> 4, c4 = idx & 15;
      async_ld_b128(lbase + (uint32_t)(r * XPAD + c4 * 4) * 4u,
                    xin + (long)r * rowStride + c4 * 4);
    }
  };

  v8f h[4];
#pragma unroll
  for (int t = 0; t < 4; ++t) h[t] = splat8(0.f);

  issue_x(0, wx0);
  for (int step = 0; step < Tsteps; ++step) {
    wait_async0();  // x tile for this step is resident in LDS
    float* wx = (step & 1) ? wx1 : wx0;
    if (step + 1 < Tsteps) issue_x(step + 1, (step & 1) ? wx0 : wx1);

    float* hrd = (step & 1) ? wh1 : wh0;
    float* hwr = (step & 1) ? wh0 : wh1;
#pragma unroll
    for (int t = 0; t < 4; ++t) {
      const int oc = t * 16 + p;
      v8f az = splat8(sbz[oc]);
      v8f vv = splat8(sbv[oc]);
      v8f ax = splat8(sbh[oc]);
      v8f uu = splat8(sbu[oc]);
      const float* ar = wx + p * XPAD + 2 * half;
      const float* wzr = sWz + oc * WPAD + 2 * half;
      const float* wvr = sWv + oc * WPAD + 2 * half;
      const float* whr = sWh + oc * WPAD + 2 * half;
#pragma unroll
      for (int kc = 0; kc < 16; ++kc) {
        v2f a = ldfrag(ar + 4 * kc);
        az = wmma4(a, ldfrag(wzr + 4 * kc), az);
        vv = wmma4(a, ldfrag(wvr + 4 * kc), vv);
        ax = wmma4(a, ldfrag(whr + 4 * kc), ax);
      }
      const float* hr = hrd + p * XPAD + 2 * half;
      const float* uzr = sUz + oc * WPAD + 2 * half;
      const float* uhr = sUh + oc * WPAD + 2 * half;
#pragma unroll
      for (int kc = 0; kc < 16; ++kc) {
        v2f a = ldfrag(hr + 4 * kc);
        az = wmma4(a, ldfrag(uzr + 4 * kc), az);
        uu = wmma4(a, ldfrag(uhr + 4 * kc), uu);
      }
#pragma unroll
      for (int j = 0; j < 8; ++j) {
        float z = 1.f / (1.f + __expf(-az[j]));
        float cand = tanhf(ax[j] + vv[j] * uu[j]);
        float hn = (1.f - z) * h[t][j] + z * cand;
        h[t][j] = hn;
        hwr[(j + 8 * half) * XPAD + oc] = hn;
      }
    }
    if (seq_out) {
      float* dst = seq_out + baseRow * (long)Tsteps * 64 + (long)step * 64;
#pragma unroll
      for (int i = 0; i < 8; ++i) {
        int idx = lane + 32 * i;
        int r = idx >> 4, c4 = idx & 15;
        f4 v = *(const f4*)(hwr + r * XPAD + c4 * 4);
        *(f4*)(dst + (long)r * Tsteps * 64 + c4 * 4) = v;
      }
    }
    if (final_out && step == Tsteps - 1) {
#pragma unroll
      for (int i = 0; i < 8; ++i) {
        int idx = lane + 32 * i;
        int r = idx >> 4, c4 = idx & 15;
        f4 v = *(const f4*)(hwr + r * XPAD + c4 * 4);
        *(f4*)(final_out + (baseRow + r) * 64 + c4 * 4) = v;
      }
    }
  }
}

// ---------------------------------------------------------------------------
// Generic WMMA GEMM: out(rows,64) = concat(X0,X1)(rows,Cin) @ W(64,Cin)^T + bias
// (X1 supplies cols 64..127 when Cin==128). 4 waves/block, 64 rows/block.
// Synchronous staging here: rows may not be a multiple of 64 (K/V: rows=32),
// and async copies cannot mask out-of-range lanes.
// ---------------------------------------------------------------------------
__global__ void __launch_bounds__(128, 1) gemm_rows_kernel(
    const float* __restrict__ X0, long ld0,
    const float* __restrict__ X1, long ld1,
    const float* __restrict__ W, const float* __restrict__ bias,
    float* __restrict__ out, int rows, int Cin) {
  extern __shared__ float smem[];
  const int WP = Cin + 2;
  float* sW = smem;
  float* sB = sW + 64 * WP;
  float* sX = sB + 64;
  const int tid = threadIdx.x;
  for (int i = tid; i < 64 * Cin; i += blockDim.x) {
    int o = i / Cin, k = i - o * Cin;
    sW[o * WP + k] = W[i];
  }
  for (int i = tid; i < 64; i += blockDim.x) sB[i] = bias ? bias[i] : 0.f;
  const int wv = tid >> 5, lane = tid & 31;
  float* wx = sX + wv * 16 * XPAD;
  __syncthreads();
  const int p = lane & 15, half = lane >> 4;
  const long baseRow = (long)blockIdx.x * 64 + wv * 16;
  v8f acc[4];
#pragma unroll
  for (int t = 0; t < 4; ++t) acc[t] = splat8(sB[t * 16 + p]);
  const int chunks = Cin >> 6;
  for (int cb = 0; cb < chunks; ++cb) {
    const float* Xp = cb ? X1 : X0;
    const long ld = cb ? ld1 : ld0;
#pragma unroll
    for (int i = 0; i < 8; ++i) {
      int idx = lane + 32 * i;
      int r = idx >> 4, c4 = idx & 15;
      long rr = baseRow + r;
      f4 v;
      if (rr < rows) v = *(const f4*)(Xp + rr * ld + c4 * 4);
      else { v[0] = 0.f; v[1] = 0.f; v[2] = 0.f; v[3] = 0.f; }
      *(f4*)(wx + r * XPAD + c4 * 4) = v;
    }
#pragma unroll
    for (int t = 0; t < 4; ++t) {
      const float* ar = wx + p * XPAD + 2 * half;
      const float* br = sW + (t * 16 + p) * WP + cb * 64 + 2 * half;
#pragma unroll
      for (int kc = 0; kc < 16; ++kc)
        acc[t] = wmma4(ldfrag(ar + 4 * kc), ldfrag(br + 4 * kc), acc[t]);
    }
  }
#pragma unroll
  for (int t = 0; t < 4; ++t)
#pragma unroll
    for (int j = 0; j < 8; ++j) {
      long row = baseRow + j + 8 * half;
      if (row < rows) out[row * 64 + t * 16 + p] = acc[t][j];
    }
}

// ---------------------------------------------------------------------------
// Strided conv1d (F->F, K=5, stride 2, pad 2). One block per stock, x row in LDS.
// Output layout (N, T2, F).
// ---------------------------------------------------------------------------
__global__ void conv_slow_kernel(const float* __restrict__ x,
                                 const float* __restrict__ cw,
                                 const float* __restrict__ cb,
                                 float* __restrict__ xs) {
  __shared__ float sx[T_ * F_];
  const int n = blockIdx.x, tid = threadIdx.x;
  const float* xrow = x + (long)n * T_ * F_;
  for (int i = tid; i < T_ * F_; i += blockDim.x) sx[i] = xrow[i];
  __syncthreads();
  const int tp = tid;  // 0..63 output time index
  for (int o = 0; o < F_; ++o) {
    float acc = cb[o];
    for (int k = 0; k < KC_; ++k) {
      int ti = 2 * tp + k - 2;
      if (ti < 0 || ti >= T_) continue;
      const float* xr = sx + ti * F_;
      for (int i = 0; i < F_; ++i) acc += xr[i] * cw[(o * F_ + i) * KC_ + k];
    }
    xs[((long)n * T2_ + tp) * F_ + o] = acc;
  }
}

// ---------------------------------------------------------------------------
// MHCA core: per-row 4-head attention against 32 latent slots (K/V in LDS).
// ---------------------------------------------------------------------------
__global__ void mhca_attn(const float* __restrict__ Q, const float* __restrict__ Km,
                          const float* __restrict__ Vm, float* __restrict__ O, int n) {
  __shared__ float sk[L_ * D_], sv[L_ * D_];
  const int tid = threadIdx.x;
  for (int i = tid; i < L_ * D_; i += blockDim.x) { sk[i] = Km[i]; sv[i] = Vm[i]; }
  __syncthreads();
  const int r = blockIdx.x * blockDim.x + tid;
  if (r >= n) return;
  for (int h = 0; h < NH_; ++h) {
    float q[HD_];
#pragma unroll
    for (int d = 0; d < HD_; ++d) q[d] = Q[(long)r * D_ + h * HD_ + d];
    float s[L_];
    float mx = -1e30f;
    for (int l = 0; l < L_; ++l) {
      float a = 0.f;
#pragma unroll
      for (int d = 0; d < HD_; ++d) a += q[d] * sk[l * D_ + h * HD_ + d];
      s[l] = a * 0.25f;  // 1/sqrt(16)
      mx = fmaxf(mx, s[l]);
    }
    float sum = 0.f;
    for (int l = 0; l < L_; ++l) { s[l] = __expf(s[l] - mx); sum += s[l]; }
    float inv = 1.f / sum;
    float o[HD_];
#pragma unroll
    for (int d = 0; d < HD_; ++d) o[d] = 0.f;
    for (int l = 0; l < L_; ++l)
#pragma unroll
      for (int d = 0; d < HD_; ++d) o[d] += s[l] * sv[l * D_ + h * HD_ + d];
#pragma unroll
    for (int d = 0; d < HD_; ++d) O[(long)r * D_ + h * HD_ + d] = o[d] * inv;
  }
}

// ---------------------------------------------------------------------------
// GAT helpers
// ---------------------------------------------------------------------------
__device__ inline unsigned encf(float f) {
  unsigned u = __float_as_uint(f);
  return (u & 0x80000000u) ? ~u : (u | 0x80000000u);
}
__device__ inline float decf(unsigned u) {
  unsigned v = (u & 0x80000000u) ? (u & 0x7fffffffu) : ~u;
  return __uint_as_float(v);
}

__global__ void rowdot_kernel(const float* __restrict__ X, int ld,
                              const float* __restrict__ w, float* __restrict__ out,
                              int n, int c) {
  int r = blockIdx.x * blockDim.x + threadIdx.x;
  if (r >= n) return;
  float acc = 0.f;
  for (int i = 0; i < c; ++i) acc += X[(long)r * ld + i] * w[i];
  out[r] = acc;
}

__global__ void dot_small(const float* a, const float* b, float* out, int c) {
  if (threadIdx.x == 0 && blockIdx.x == 0) {
    float s = 0.f;
    for (int i = 0; i < c; ++i) s += a[i] * b[i];
    out[0] = s;
  }
}

__global__ void fill_u32(uint32_t* p, uint32_t v, int n) {
  int i = blockIdx.x * blockDim.x + threadIdx.x;
  if (i < n) p[i] = v;
}

__global__ void gat_edge1(const int* __restrict__ src, const int* __restrict__ dst,
                          const float* __restrict__ ew, const float* __restrict__ asrc,
                          const float* __restrict__ adst, const float* __restrict__ coef,
                          float* __restrict__ alpha, uint32_t* __restrict__ mEnc, int e) {
  int i = blockIdx.x * blockDim.x + threadIdx.x;
  if (i >= e) return;
  int s = src[i], d = dst[i];
  float a = asrc[s] + adst[d] + coef[0] * ew[i];
  a = a > 0.f ? a : 0.2f * a;  // leaky_relu(0.2)
  alpha[i] = a;
  atomicMax((unsigned int*)(mEnc + d), encf(a));
}

__global__ void gat_edge2(const int* __restrict__ dst, const float* __restrict__ alpha,
                          const uint32_t* __restrict__ mEnc, float* __restrict__ ex,
                          float* __restrict__ den, int e) {
  int i = blockIdx.x * blockDim.x + threadIdx.x;
  if (i >= e) return;
  int d = dst[i];
  float v = __expf(alpha[i] - decf(mEnc[d]));
  ex[i] = v;
  atomicAdd(den + d, v);
}

__global__ void gat_scatter(const int* __restrict__ src, const int* __restrict__ dst,
                            const float* __restrict__ ex, const float* __restrict__ den,
                            const float* __restrict__ xw, float* __restrict__ out,
                            int e, int c) {
  long gi = (long)blockIdx.x * blockDim.x + threadIdx.x;
  if (gi >= (long)e * c) return;
  int i = (int)(gi / c), j = (int)(gi - (long)i * c);
  int s = src[i], d = dst[i];
  float w = ex[i] / (den[d] + 1e-16f);
  atomicAdd(out + (long)d * c + j, xw[(long)s * c + j] * w);
}

__global__ void bias_act(const float* __restrict__ in, const float* __restrict__ bias,
                         float* __restrict__ out, int n, int c, int elu) {
  long gi = (long)blockIdx.x * blockDim.x + threadIdx.x;
  if (gi >= (long)n * c) return;
  int j = (int)(gi % c);
  float v = in[gi] + bias[j];
  if (elu) v = v > 0.f ? v : (__expf(v) - 1.f);
  out[gi] = v;
}

// ---------------------------------------------------------------------------
// Host: param flattening assumes top-level setup_inputs() order (x, edge_index,
// edge_weight, params) with params leaves in jax tree order (dict keys sorted
// recursively). Indices documented so they can be remapped in one place.
// ---------------------------------------------------------------------------
constexpr int IN_X = 0, IN_EI = 1, IN_EW = 2;
constexpr int P_COMB_B = 3, P_COMB_W = 4, P_CONV_B = 5, P_CONV_W = 6;
constexpr int P_FAST0 = 7, P_FAST1 = 21;
constexpr int P_GATCS1 = 35, P_GATCS2 = 41, P_GATP1 = 47, P_GATP2 = 53;
constexpr int P_LAT = 59;
constexpr int P_SLOW0 = 77, P_SLOW1 = 91;
// GRU block (sorted): +0 Uh_b +1 Uh_w +2 Uz_b +3 Uz_w +4 Wh_b +5 Wh_w +6 Wk_b
//   +7 Wk_w +8 Wq_b +9 Wq_w +10 Wv_b +11 Wv_w +12 Wz_b +13 Wz_w
// GAT block (sorted): +0 att_dst +1 att_edge +2 att_src +3 bias +4 lin_edge_w +5 lin_w
// LAT block (sorted): +0 R1 +1 R2 +2 WK1_b +3 WK1_w +4 WK2_b +5 WK2_w +6 WO1_b
//   +7 WO1_w +8 WO2_b +9 WO2_w +10 WQ1_b +11 WQ1_w +12 WQ2_b +13 WQ2_w
//   +14 WV1_b +15 WV1_w +16 WV2_b +17 WV2_w

extern "C" void kernel_launch(void* const* d_in, const int* in_sizes, int n_in,
                              void* d_out, int out_size, void* d_ws, size_t ws_size,
                              hipStream_t stream) {
  (void)in_sizes; (void)n_in; (void)out_size; (void)ws_size;
  auto Pf = [&](int i) { return (const float*)d_in[i]; };
  const float* x = Pf(IN_X);
  const int* ei = (const int*)d_in[IN_EI];
  const float* ew = Pf(IN_EW);
  const int* src = ei;
  const int* dst = ei + E_;

  // ---- workspace carve ----
  float* base = (float*)d_ws;
  size_t off = 0;
  auto alloc = [&](size_t n) { float* p = base + off; off += n; return p; };
  float* xs = alloc((size_t)N_ * T2_ * F_);
  float* H1f = alloc((size_t)N_ * T_ * D_);
  float* H1s = alloc((size_t)N_ * T2_ * D_);
  float* fastF = alloc((size_t)N_ * D_);
  float* slowF = alloc((size_t)N_ * D_);
  float* A1 = alloc((size_t)N_ * D_);
  float* A2 = alloc((size_t)N_ * D_);
  float* xw = alloc((size_t)N_ * D_);
  float* hbuf = alloc((size_t)N_ * D_);
  float* agg = alloc((size_t)N_ * D_);
  float* Qb = alloc((size_t)N_ * D_);
  float* Kb = alloc((size_t)L_ * D_);
  float* Vb = alloc((size_t)L_ * D_);
  float* attO = alloc((size_t)N_ * D_);
  float* B1 = alloc((size_t)N_ * D_);
  float* B2 = alloc((size_t)N_ * D_);
  float* asrc = alloc(N_);
  float* adst = alloc(N_);
  float* coef = alloc(64);
  uint32_t* mEnc = (uint32_t*)alloc(N_);
  float* den = alloc(N_);
  float* alphaE = alloc(E_);
  float* exE = alloc(E_);
  float* xwp2 = alloc(N_);
  float* aggp = alloc(N_);

  const int gruSm = (5 * 64 * WPAD + 4 * 64 + 8 * 16 * XPAD + 8 * 16 * XPAD) * 4;
  const int gemmSmMax = (64 * 130 + 64 + 4 * 16 * XPAD) * 4;
  hipFuncSetAttribute((const void*)gru_layer_kernel,
                      hipFuncAttributeMaxDynamicSharedMemorySize, gruSm);
  hipFuncSetAttribute((const void*)gemm_rows_kernel,
                      hipFuncAttributeMaxDynamicSharedMemorySize, gemmSmMax);

  // ---- multi-scale temporal encoder ----
  conv_slow_kernel<<<N_, 64, 0, stream>>>(x, Pf(P_CONV_W), Pf(P_CONV_B), xs);

  auto gru = [&](const float* in, long rs, long ts, int Tn, int pb, float* seq, float* fin) {
    gru_layer_kernel<<<N_ / 64, 128, gruSm, stream>>>(
        in, rs, ts, Tn,
        Pf(pb + 13), Pf(pb + 12),  // Wz
        Pf(pb + 3), Pf(pb + 2),    // Uz
        Pf(pb + 11), Pf(pb + 10),  // Wv
        Pf(pb + 5), Pf(pb + 4),    // Wh
        Pf(pb + 1), Pf(pb + 0),    // Uh
        seq, fin);
  };
  gru(x, (long)T_ * F_, F_, T_, P_FAST0, H1f, nullptr);
  gru(H1f, (long)T_ * D_, D_, T_, P_FAST1, nullptr, fastF);
  gru(xs, (long)T2_ * F_, F_, T2_, P_SLOW0, H1s, nullptr);
  gru(H1s, (long)T2_ * D_, D_, T2_, P_SLOW1, nullptr, slowF);

  auto gemm = [&](const float* X0, long ld0, const float* X1, long ld1,
                  const float* W, const float* bias, float* outp, int rows, int Cin) {
    int sm = (64 * (Cin + 2) + 64 + 4 * 16 * XPAD) * 4;
    gemm_rows_kernel<<<(rows + 63) / 64, 128, sm, stream>>>(X0, ld0, X1, ld1, W, bias,
                                                            outp, rows, Cin);
  };
  gemm(fastF, 64, slowF, 64, Pf(P_COMB_W), Pf(P_COMB_B), A1, N_, 128);  // comb

  // ---- GAT conv with cout=64 (heads=1; mean over 1 head == identity) ----
  auto gat64 = [&](const float* Xa, long lda, const float* Xb, long ldb, int Cin,
                   int pb, float* outp, int elu) {
    gemm(Xa, lda, Xb, ldb, Pf(pb + 5), nullptr, xw, N_, Cin);
    rowdot_kernel<<<(N_ + 255) / 256, 256, 0, stream>>>(xw, 64, Pf(pb + 2), asrc, N_, 64);
    rowdot_kernel<<<(N_ + 255) / 256, 256, 0, stream>>>(xw, 64, Pf(pb + 0), adst, N_, 64);
    dot_small<<<1, 1, 0, stream>>>(Pf(pb + 4), Pf(pb + 1), coef, 64);
    fill_u32<<<(N_ + 255) / 256, 256, 0, stream>>>(mEnc, 0x007FFFFFu, N_);  // enc(-inf)
    fill_u32<<<(N_ + 255) / 256, 256, 0, stream>>>((uint32_t*)den, 0u, N_);
    fill_u32<<<(N_ * 64 + 255) / 256, 256, 0, stream>>>((uint32_t*)agg, 0u, N_ * 64);
    gat_edge1<<<(E_ + 255) / 256, 256, 0, stream>>>(src, dst, ew, asrc, adst, coef,
                                                    alphaE, mEnc, E_);
    gat_edge2<<<(E_ + 255) / 256, 256, 0, stream>>>(dst, alphaE, mEnc, exE, den, E_);
    gat_scatter<<<(unsigned)(((long)E_ * 64 + 255) / 256), 256, 0, stream>>>(
        src, dst, exE, den, xw, agg, E_, 64);
    bias_act<<<(N_ * 64 + 255) / 256, 256, 0, stream>>>(agg, Pf(pb + 3), outp, N_, 64, elu);
  };

  // cross-sectional GAT on last-day features x[0,:,-1,:]
  gat64(x + (T_ - 1) * F_, (long)T_ * F_, nullptr, 0, 64, P_GATCS1, hbuf, 1);
  gat64(hbuf, 64, nullptr, 0, 64, P_GATCS2, A2, 0);

  // ---- latent cross-attention (suffix 1 and 2) ----
  auto mhca = [&](const float* Aq, const float* R, int qw, int qb, int kw, int kb,
                  int vw, int vb, int ow, int ob, float* outp) {
    gemm(Aq, 64, nullptr, 0, Pf(qw), Pf(qb), Qb, N_, 64);
    gemm(R, 64, nullptr, 0, Pf(kw), Pf(kb), Kb, L_, 64);
    gemm(R, 64, nullptr, 0, Pf(vw), Pf(vb), Vb, L_, 64);
    mhca_attn<<<(N_ + 127) / 128, 128, 0, stream>>>(Qb, Kb, Vb, attO, N_);
    gemm(attO, 64, nullptr, 0, Pf(ow), Pf(ob), outp, N_, 64);
  };
  mhca(A1, Pf(P_LAT + 0), P_LAT + 11, P_LAT + 10, P_LAT + 3, P_LAT + 2,
       P_LAT + 15, P_LAT + 14, P_LAT + 7, P_LAT + 6, B1);
  mhca(A2, Pf(P_LAT + 1), P_LAT + 13, P_LAT + 12, P_LAT + 5, P_LAT + 4,
       P_LAT + 17, P_LAT + 16, P_LAT + 9, P_LAT + 8, B2);

  // ---- prediction GAT: conv1 on concat(B1,B2), conv2 cout=1 ----
  gat64(B1, 64, B2, 64, 128, P_GATP1, hbuf, 1);

  rowdot_kernel<<<(N_ + 255) / 256, 256, 0, stream>>>(hbuf, 64, Pf(P_GATP2 + 5), xwp2, N_, 64);
  rowdot_kernel<<<(N_ + 255) / 256, 256, 0, stream>>>(xwp2, 1, Pf(P_GATP2 + 2), asrc, N_, 1);
  rowdot_kernel<<<(N_ + 255) / 256, 256, 0, stream>>>(xwp2, 1, Pf(P_GATP2 + 0), adst, N_, 1);
  dot_small<<<1, 1, 0, stream>>>(Pf(P_GATP2 + 4), Pf(P_GATP2 + 1), coef, 1);
  fill_u32<<<(N_ + 255) / 256, 256, 0, stream>>>(mEnc, 0x007FFFFFu, N_);
  fill_u32<<<(N_ + 255) / 256, 256, 0, stream>>>((uint32_t*)den, 0u, N_);
  fill_u32<<<(N_ + 255) / 256, 256, 0, stream>>>((uint32_t*)aggp, 0u, N_);
  gat_edge1<<<(E_ + 255) / 256, 256, 0, stream>>>(src, dst, ew, asrc, adst, coef,
                                                  alphaE, mEnc, E_);
  gat_edge2<<<(E_ + 255) / 256, 256, 0, stream>>>(dst, alphaE, mEnc, exE, den, E_);
  gat_scatter<<<(E_ + 255) / 256, 256, 0, stream>>>(src, dst, exE, den, xwp2, aggp, E_, 1);
  bias_act<<<(N_ + 255) / 256, 256, 0, stream>>>(aggp, Pf(P_GATP2 + 3), (float*)d_out,
                                                 N_, 1, 0);
}